// DynaKeyMemoryCore_79456894976557
// MI455X (gfx1250) — compile-verified
//
#include <hip/hip_runtime.h>
#include <hip/hip_bf16.h>
#include <math.h>

// Problem constants (match reference)
#define Bq  4
#define Nq  8
#define Cq  256
#define Hq  64
#define Wq  64
#define Kq  4
#define HIDq 256
#define NAq 5
#define BN  (Bq*Nq)              // 32
#define PIX (Hq*Wq)              // 4096
#define DIN (3*Cq + Kq)          // 772 (multiple of 4)
#define DTq 1.0f

typedef __attribute__((ext_vector_type(2))) float v2f;
typedef __attribute__((ext_vector_type(8))) float v8f;

#define WMMA_F32(a, b, c) \
    __builtin_amdgcn_wmma_f32_16x16x4_f32(false, (a), false, (b), (short)0, (c), false, false)

// ---- workspace layout (in floats) ----
#define WS_S1    0                     // [BN*C]  sum(v*m)
#define WS_S0    (WS_S1 + BN*Cq)       // [BN*C]  sum(v)
#define WS_DEN   (WS_S0 + BN*Cq)       // [BN]
#define WS_Z     (WS_DEN + BN)         // [BN*C]
#define WS_DELTA (WS_Z + BN*Cq)        // [BN*C]  (z_next - z)
#define WS_WB    (WS_DELTA + BN*Cq)    // [BN*K]  retrieval weights
#define WS_H     (WS_WB + BN*Kq)       // [BN*HID]
#define WS_QB    (WS_H + BN*HIDq)      // [BN*NA]
#define WS_BF    (WS_QB + BN*NAq)      // [BN] int: bank_full
#define WS_ACT   (WS_BF + BN)          // [BN] int: action

// ---- output layout (floats, concatenated in return order) ----
#define OUT_READ 0
#define OUT_W    (Bq*Nq*Cq*Hq*Wq)          // 33554432
#define OUT_Q    (OUT_W + BN*Kq)           // +128
#define OUT_ACT  (OUT_Q + BN*NAq)          // +160
#define OUT_ENT  (OUT_ACT + BN)            // +32
#define OUT_HIST (OUT_ENT + BN)            // +32

// --------------------------------------------------------------------------
// Kernel 1: denom[bn] = sum over HxW of mask
__global__ void dkm_mask_denom(const float* __restrict__ mask, float* __restrict__ ws) {
    int bn = blockIdx.x;
    const float* m = mask + (size_t)bn * PIX;
    float s = 0.f;
    for (int i = threadIdx.x; i < PIX; i += blockDim.x) s += m[i];
    for (int off = 16; off > 0; off >>= 1) s += __shfl_down(s, off, 32);
    __shared__ float red[8];
    int wid = threadIdx.x >> 5, lane = threadIdx.x & 31;
    if (lane == 0) red[wid] = s;
    __syncthreads();
    if (threadIdx.x == 0) {
        float t = 0.f;
        int nw = blockDim.x >> 5;
        for (int i = 0; i < nw; ++i) t += red[i];
        ws[WS_DEN + bn] = t;
    }
}

// --------------------------------------------------------------------------
// Kernel 2: per (b,n,c): s1 = sum v*m, s0 = sum v   (the 128MB streaming read)
__global__ void dkm_pool_reduce(const float* __restrict__ value,
                                const float* __restrict__ mask,
                                float* __restrict__ ws) {
    int bid = blockIdx.x;            // b*N*C + n*C + c
    int bn  = bid / Cq;
    const float* v = value + (size_t)bid * PIX;
    const float* m = mask  + (size_t)bn  * PIX;
    float a = 0.f, b = 0.f;
    for (int i = threadIdx.x; i < PIX; i += blockDim.x) {
        float vv = v[i];
        a += vv * m[i];
        b += vv;
    }
    for (int off = 16; off > 0; off >>= 1) {
        a += __shfl_down(a, off, 32);
        b += __shfl_down(b, off, 32);
    }
    __shared__ float ra[8], rb[8];
    int wid = threadIdx.x >> 5, lane = threadIdx.x & 31;
    if (lane == 0) { ra[wid] = a; rb[wid] = b; }
    __syncthreads();
    if (threadIdx.x == 0) {
        float ta = 0.f, tb = 0.f;
        int nw = blockDim.x >> 5;
        for (int i = 0; i < 8; ++i) { ta += ra[i]; tb += rb[i]; }
        (void)nw;
        ws[WS_S1 + bid] = ta;
        ws[WS_S0 + bid] = tb;
    }
}

// --------------------------------------------------------------------------
// Kernel 3: the whole "brain" in one 1024-thread (32-wave) block.
__global__ void __launch_bounds__(1024)
dkm_brain(const float* __restrict__ dict_keys,
          const float* __restrict__ dict_vel,
          const unsigned char* __restrict__ dict_valid,
          const float* __restrict__ W1, const float* __restrict__ b1,
          const float* __restrict__ W2, const float* __restrict__ b2,
          float* __restrict__ ws, float* __restrict__ out) {
    const int tid  = threadIdx.x;
    const int wave = tid >> 5;
    const int lane = tid & 31;

    float* z   = ws + WS_Z;
    float* dl  = ws + WS_DELTA;
    float* wb  = ws + WS_WB;
    float* hwb = ws + WS_H;
    float* qb  = ws + WS_QB;
    int*   bf  = (int*)(ws + WS_BF);
    int*   act = (int*)(ws + WS_ACT);

    // Zero-padded W2 (256 x 16) staged in LDS for branch-free B-fragments.
    __shared__ float w2p[HIDq * 16];

    // ---- Phase A: z selection + W2 staging -------------------------------
    for (int i = tid; i < BN * Cq; i += 1024) {
        int bn = i / Cq;
        float den = ws[WS_DEN + bn];
        float pooled   = ws[WS_S1 + i] / fmaxf(den, 1e-6f);
        float fallback = ws[WS_S0 + i] * (1.0f / (float)PIX);
        z[i] = (den > 1e-5f) ? pooled : fallback;
    }
    for (int i = tid; i < HIDq * 16; i += 1024) {
        int k = i >> 4, n = i & 15;
        w2p[i] = (n < NAq) ? W2[k * NAq + n] : 0.f;
    }
    __threadfence();
    __syncthreads();

    // ---- Phase B: retrieval / spawn / re-retrieval / delta  (wave = bn) --
    {
        int bn = wave;
        float dist[Kq];
        for (int k = 0; k < Kq; ++k) {
            const float* key = dict_keys + ((size_t)bn * Kq + k) * Cq;
            float p = 0.f;
            for (int c = lane; c < Cq; c += 32) {
                float d = z[bn * Cq + c] - key[c];
                p += d * d;
            }
            for (int off = 16; off > 0; off >>= 1) p += __shfl_down(p, off, 32);
            dist[k] = __shfl(p, 0, 32);            // broadcast to all lanes
        }
        bool valid[Kq];
        for (int k = 0; k < Kq; ++k) valid[k] = dict_valid[bn * Kq + k] != 0;

        bool has0 = valid[0] | valid[1] | valid[2] | valid[3];
        int slot = 0;                                // first invalid slot
        for (int k = Kq - 1; k >= 0; --k) if (!valid[k]) slot = k;
        bool need = !has0;
        bool wr[Kq];
        bool valid2[Kq];
        float dist2[Kq];
        for (int k = 0; k < Kq; ++k) {
            wr[k]     = need && (k == slot);
            valid2[k] = valid[k] || wr[k];
            dist2[k]  = wr[k] ? 0.f : dist[k];
        }
        bool has = valid2[0] | valid2[1] | valid2[2] | valid2[3];
        float lg[Kq], mx = -1e30f;
        for (int k = 0; k < Kq; ++k) { lg[k] = valid2[k] ? -dist2[k] : -1e30f; mx = fmaxf(mx, lg[k]); }
        float es = 0.f, wv[Kq];
        for (int k = 0; k < Kq; ++k) { wv[k] = __expf(lg[k] - mx); es += wv[k]; }
        for (int k = 0; k < Kq; ++k) wv[k] = has ? (wv[k] / es) : 0.f;

        bool bankfull = valid2[0] & valid2[1] & valid2[2] & valid2[3];
        float ent = 0.f;
        for (int k = 0; k < Kq; ++k) {
            float wc = fmaxf(wv[k], 1e-8f);
            ent -= wc * __logf(wc);
        }
        if (lane == 0) {
            for (int k = 0; k < Kq; ++k) {
                wb[bn * Kq + k]          = wv[k];
                out[OUT_W + bn * Kq + k] = wv[k];
            }
            out[OUT_ENT + bn] = ent;
            bf[bn] = bankfull ? 1 : 0;
        }
        for (int c = lane; c < Cq; c += 32) {
            float vbar = 0.f;
            for (int k = 0; k < Kq; ++k) {
                float vel = wr[k] ? 0.f : dict_vel[((size_t)bn * Kq + k) * Cq + c];
                vbar += wv[k] * vel;
            }
            dl[bn * Cq + c] = DTq * vbar;
        }
    }
    __threadfence();
    __syncthreads();

    // ---- Phase C: h = relu(q_state @ W1 + b1) via V_WMMA_F32_16X16X4_F32 -
    // q_state = [z | z+delta | delta | w]; processed as 4 branch-free K segments.
    {
        const int Mb   = (wave >> 4) * 16;  // 0 or 16
        const int Nb   = (wave & 15) * 16;  // 0..240
        const int half = lane >> 4;         // K sub-pair select (ISA A layout)
        const int l16  = lane & 15;
        const int m    = Mb + l16;
        const int nc   = Nb + l16;
        v8f acc = {};

        // Segment 1: cols [0,256) = z
        #pragma unroll 4
        for (int kk = 0; kk < Cq; kk += 4) {
            int k0 = kk + half * 2;                    // even
            v2f a = *(const v2f*)&z[m * Cq + k0];
            v2f b;
            b.x = W1[(size_t)k0 * HIDq + nc];
            b.y = W1[(size_t)(k0 + 1) * HIDq + nc];
            acc = WMMA_F32(a, b, acc);
        }
        // Segment 2: cols [256,512) = z + delta
        #pragma unroll 4
        for (int kk = 0; kk < Cq; kk += 4) {
            int k0 = kk + half * 2;
            v2f az = *(const v2f*)&z[m * Cq + k0];
            v2f ad = *(const v2f*)&dl[m * Cq + k0];
            v2f a = az + ad;
            v2f b;
            b.x = W1[(size_t)(Cq + k0) * HIDq + nc];
            b.y = W1[(size_t)(Cq + k0 + 1) * HIDq + nc];
            acc = WMMA_F32(a, b, acc);
        }
        // Segment 3: cols [512,768) = delta
        #pragma unroll 4
        for (int kk = 0; kk < Cq; kk += 4) {
            int k0 = kk + half * 2;
            v2f a = *(const v2f*)&dl[m * Cq + k0];
            v2f b;
            b.x = W1[(size_t)(2 * Cq + k0) * HIDq + nc];
            b.y = W1[(size_t)(2 * Cq + k0 + 1) * HIDq + nc];
            acc = WMMA_F32(a, b, acc);
        }
        // Segment 4: cols [768,772) = w
        {
            int k0 = half * 2;
            v2f a = *(const v2f*)&wb[m * Kq + k0];
            v2f b;
            b.x = W1[(size_t)(3 * Cq + k0) * HIDq + nc];
            b.y = W1[(size_t)(3 * Cq + k0 + 1) * HIDq + nc];
            acc = WMMA_F32(a, b, acc);
        }
        // C/D layout: VGPR r -> row Mb + half*8 + r, col Nb + l16
        for (int r = 0; r < 8; ++r) {
            int row = Mb + half * 8 + r;
            int col = Nb + l16;
            float v = acc[r] + b1[col];
            hwb[row * HIDq + col] = fmaxf(v, 0.f);
        }
    }
    __threadfence();
    __syncthreads();

    // ---- Phase D: q = h @ W2pad + b2 (N padded 5->16), waves 0..1 only ---
    if (wave < 2) {
        const int Mb   = wave * 16;
        const int half = lane >> 4;
        const int l16  = lane & 15;
        v8f acc = {};
        #pragma unroll 4
        for (int kk = 0; kk < HIDq; kk += 4) {
            int k0 = kk + half * 2;
            v2f a = *(const v2f*)&hwb[(Mb + l16) * HIDq + k0];
            v2f b;
            b.x = w2p[k0 * 16 + l16];
            b.y = w2p[(k0 + 1) * 16 + l16];
            acc = WMMA_F32(a, b, acc);
        }
        for (int r = 0; r < 8; ++r) {
            int row = Mb + half * 8 + r;
            if (l16 < NAq) qb[row * NAq + l16] = acc[r] + b2[l16];
        }
    }
    __threadfence();
    __syncthreads();

    // ---- Phase E: masked argmax, q_values / actions out ------------------
    if (tid < BN) {
        int bn = tid;
        bool bankfull = bf[bn] != 0;
        float best = -1e38f;
        int besta = 0;
        for (int a = 0; a < NAq; ++a) {
            float qv = qb[bn * NAq + a];
            if (a == NAq - 1 && bankfull) qv = -1e9f;
            out[OUT_Q + bn * NAq + a] = qv;
            if (qv > best) { best = qv; besta = a; }
        }
        act[bn] = besta;
        out[OUT_ACT + bn] = (float)besta;
    }
    __threadfence();
    __syncthreads();

    // ---- Phase F: action histogram ---------------------------------------
    if (tid < NAq) {
        int cnt = 0;
        for (int i = 0; i < BN; ++i) cnt += (act[i] == tid) ? 1 : 0;
        out[OUT_HIST + tid] = (float)cnt / (float)BN;
    }
}

// --------------------------------------------------------------------------
// Kernel 4: readout = value + gate * delta  (broadcast over HxW), float4
__global__ void dkm_readout(const float4* __restrict__ value,
                            const float* __restrict__ ws,
                            const float* __restrict__ gate,
                            float4* __restrict__ out) {
    size_t i = (size_t)blockIdx.x * blockDim.x + threadIdx.x;  // over 8,388,608 float4
    float g = gate[0];
    int bnc = (int)(i >> 10);          // 1024 float4 per (b,n,c) row
    float d = g * ws[WS_DELTA + bnc];
    float4 v = value[i];
    v.x += d; v.y += d; v.z += d; v.w += d;
    out[i] = v;
}

// --------------------------------------------------------------------------
extern "C" void kernel_launch(void* const* d_in, const int* in_sizes, int n_in,
                              void* d_out, int out_size, void* d_ws, size_t ws_size,
                              hipStream_t stream) {
    const float* value = (const float*)d_in[0];
    const float* mask  = (const float*)d_in[1];
    const float* keys  = (const float*)d_in[2];
    const float* vel   = (const float*)d_in[3];
    const unsigned char* dvalid = (const unsigned char*)d_in[4];
    const float* W1    = (const float*)d_in[5];
    const float* b1    = (const float*)d_in[6];
    const float* W2    = (const float*)d_in[7];
    const float* b2    = (const float*)d_in[8];
    const float* gate  = (const float*)d_in[9];

    float* out = (float*)d_out;
    float* ws  = (float*)d_ws;

    // 1) mask denominators
    dkm_mask_denom<<<BN, 256, 0, stream>>>(mask, ws);
    // 2) masked channel pooling (128MB streaming read)
    dkm_pool_reduce<<<BN * Cq, 256, 0, stream>>>(value, mask, ws);
    // 3) retrieval + ODE step + WMMA MLP + actions/entropy/hist
    dkm_brain<<<1, 1024, 0, stream>>>(keys, vel, dvalid, W1, b1, W2, b2, ws, out);
    // 4) broadcast readout (256MB pass; value likely resident in 192MB L2)
    const size_t nvec4 = (size_t)Bq * Nq * Cq * Hq * Wq / 4;   // 8,388,608
    dkm_readout<<<(unsigned)(nvec4 / 256), 256, 0, stream>>>(
        (const float4*)value, ws, gate, (float4*)out /* OUT_READ == 0 */);
}